// BEVPoolV2_83245056131101
// MI455X (gfx1250) — compile-verified
//
#include <hip/hip_runtime.h>
#include <hip/hip_bf16.h>

// BEVPoolV2 scatter-accumulate for MI455X (gfx1250, wave32).
// Memory-bound: ~370 MB moved, 0.16 GFLOP. Strategy:
//  - b128 (float4) non-temporal loads for the streamed feat rows
//  - NT hints on single-use streams so the 12.8 MB accumulator stays L2-resident
//  - hardware global_atomic_add_f32 (no-return) for the scatter

typedef float vfloat4 __attribute__((ext_vector_type(4)));

#define BEV_C 80
#define BEV_VEC 4
#define BEV_CHUNKS (BEV_C / BEV_VEC)  // 20 float4 chunks per point

__device__ __forceinline__ void hw_atomic_add_f32(float* p, float v) {
#if defined(__has_builtin) && __has_builtin(__builtin_amdgcn_global_atomic_fadd_f32)
    __builtin_amdgcn_global_atomic_fadd_f32(
        (__attribute__((address_space(1))) float*)p, v);
#else
    // HIP unsafe (hardware) float atomic add; lowers to global_atomic_add_f32
    unsafeAtomicAdd(p, v);
#endif
}

__global__ __launch_bounds__(256) void bevpool_zero_kernel(vfloat4* __restrict__ out,
                                                           int n4) {
    int idx = blockIdx.x * blockDim.x + threadIdx.x;
    if (idx < n4) {
        out[idx] = (vfloat4){0.f, 0.f, 0.f, 0.f};
    }
}

__global__ __launch_bounds__(256) void bevpool_scatter_kernel(
    const int* __restrict__ ranks_depth,
    const int* __restrict__ ranks_feat,
    const int* __restrict__ ranks_bev,
    const int* __restrict__ n_points_ptr,
    const float* __restrict__ depth,
    const float* __restrict__ feat,
    float* __restrict__ out,
    unsigned total_chunks)  // P * BEV_CHUNKS (upper bound on work)
{
    unsigned idx = blockIdx.x * blockDim.x + threadIdx.x;
    if (idx >= total_chunks) return;

    unsigned i = idx / BEV_CHUNKS;  // point index
    unsigned j = idx % BEV_CHUNKS;  // float4 chunk within the 80 channels

    const int n_points = *n_points_ptr;  // uniform scalar load (L2-cached)
    if ((int)i >= n_points) return;

    // Index streams: read once, never reused -> non-temporal.
    const int rd = __builtin_nontemporal_load(ranks_depth + i);
    const int rf = __builtin_nontemporal_load(ranks_feat + i);
    const int rb = __builtin_nontemporal_load(ranks_bev + i);

    // Depth: random 4B gather into a 23.8 MB array; reused across the 20
    // chunk-lanes of this point -> regular temporal policy (L2-resident).
    const float d = depth[rd];

    // Feat row chunk: 16B b128 load, streamed once -> non-temporal so the
    // 320 MB stream does not evict the accumulator or depth from L2.
    const vfloat4* frow =
        (const vfloat4*)(feat + (size_t)rf * BEV_C) + j;
    vfloat4 f = __builtin_nontemporal_load(frow);

    f *= d;

    float* o = out + (size_t)rb * BEV_C + (size_t)j * BEV_VEC;
    hw_atomic_add_f32(o + 0, f.x);
    hw_atomic_add_f32(o + 1, f.y);
    hw_atomic_add_f32(o + 2, f.z);
    hw_atomic_add_f32(o + 3, f.w);
}

extern "C" void kernel_launch(void* const* d_in, const int* in_sizes, int n_in,
                              void* d_out, int out_size, void* d_ws, size_t ws_size,
                              hipStream_t stream) {
    const int*   ranks_depth = (const int*)d_in[0];
    const int*   ranks_feat  = (const int*)d_in[1];
    const int*   ranks_bev   = (const int*)d_in[2];
    const int*   n_points_p  = (const int*)d_in[3];  // scalar on device
    const float* depth       = (const float*)d_in[4];
    const float* feat        = (const float*)d_in[5];
    float*       out         = (float*)d_out;

    // 1) Zero the 3.2M-float accumulator (harness poisons d_out).
    int n4 = out_size / BEV_VEC;
    int zblocks = (n4 + 255) / 256;
    bevpool_zero_kernel<<<zblocks, 256, 0, stream>>>((vfloat4*)out, n4);

    // 2) Scatter-accumulate. Grid sized from P (host-known upper bound);
    //    device-side trim against *n_points_p keeps it deterministic.
    const int P = in_sizes[0];
    unsigned long long work = (unsigned long long)P * BEV_CHUNKS;
    unsigned total_chunks = (unsigned)work;
    int sblocks = (int)((work + 255ull) / 256ull);
    bevpool_scatter_kernel<<<sblocks, 256, 0, stream>>>(
        ranks_depth, ranks_feat, ranks_bev, n_points_p, depth, feat, out,
        total_chunks);
}